// Model_71837622993515
// MI455X (gfx1250) — compile-verified
//
#include <hip/hip_runtime.h>
#include <math.h>

// ---------------- problem constants ----------------
#define EMB   50
#define DEF   5
#define HID   128
#define RESH  32
#define KCB   10000
#define BATCH 4096
#define TOK   (BATCH * DEF)      // 20480 tokens
#define ZLEN  (DEF * EMB)        // 250

#define HSTR  132                // LDS row stride for 128-wide activations (bank skew)
#define USTR  36                 // LDS row stride for 32-wide residual hidden
#define CSTR  52                 // LDS row stride for staged codebook tile (K padded 50->52)

typedef __attribute__((ext_vector_type(2))) float v2f;
typedef __attribute__((ext_vector_type(8))) float v8f;

__device__ __forceinline__ v8f wmma_f32(v2f a, v2f b, v8f c) {
  // V_WMMA_F32_16X16X4_F32 : D(16x16 f32) = A(16x4 f32) x B(4x16 f32) + C
  return __builtin_amdgcn_wmma_f32_16x16x4_f32(false, a, false, b, (short)0, c, false, false);
}

__device__ __forceinline__ v8f zero8() {
  v8f z = {0.f, 0.f, 0.f, 0.f, 0.f, 0.f, 0.f, 0.f};
  return z;
}

__device__ __forceinline__ void wait_asynccnt0() {
#if __has_builtin(__builtin_amdgcn_s_wait_asynccnt)
  __builtin_amdgcn_s_wait_asynccnt(0);
#else
  asm volatile("s_wait_asynccnt 0x0" ::: "memory");
#endif
}

// Async-copy one 16x50 codebook tile (codes n0..n0+15) into LDS at byte
// offset ldsBase, laid out [16][CSTR]. 400 b64 transfers spread over the WG.
// Columns 50/51 are never written (pre-zeroed pad).
__device__ __forceinline__ void stage_tile_async(const float* __restrict__ cb,
                                                 unsigned ldsBase, int n0, int tid) {
  for (int i = tid; i < 16 * 25; i += 256) {
    const int c = i / 25;                 // code row within tile
    const int k = 2 * (i - c * 25);       // even column 0..48
    const unsigned loff = ldsBase + (unsigned)((c * CSTR + k) * 4);
    const float* g = cb + (size_t)(n0 + c) * EMB + k;
    asm volatile("global_load_async_to_lds_b64 %0, %1, off"
                 :: "v"(loff), "v"(g) : "memory");
  }
}

// =====================================================================
// Encoder: x[4096x50] -> z[4096x250] (written to workspace "zbuf")
// One WG = 256 threads (8 waves) handles 128 batch rows. Each wave owns a
// 16-row stripe; activations live in LDS (C-layout store == A-layout reload
// transpose). All GEMMs via f32 WMMA, K in steps of 4.
// =====================================================================
__global__ __launch_bounds__(256) void enc_kernel(
    const float* __restrict__ x,
    const float* __restrict__ w1,  const float* __restrict__ b1,
    const float* __restrict__ w2,  const float* __restrict__ b2,
    const float* __restrict__ w3,  const float* __restrict__ b3,
    const float* __restrict__ rw1, const float* __restrict__ rb1,
    const float* __restrict__ rw2, const float* __restrict__ rb2,
    const float* __restrict__ pw,  const float* __restrict__ pb,
    float* __restrict__ zbuf)
{
  extern __shared__ float smem[];
  float* bufA = smem;                    // 128 x HSTR
  float* bufB = smem + 128 * HSTR;       // 128 x HSTR
  float* bufU = smem + 2 * 128 * HSTR;   // 128 x USTR

  const int tid  = threadIdx.x;
  const int lane = tid & 31;
  const int wave = tid >> 5;
  const int half = lane >> 4;      // 0: lanes 0-15, 1: lanes 16-31
  const int lm   = lane & 15;
  const int row0 = blockIdx.x * 128;
  const int m0   = wave * 16;      // this wave's row stripe inside the WG

  // ---- stage x into bufA, K padded 50 -> 52 with zeros ----
  for (int i = tid; i < 128 * 52; i += 256) {
    int r = i / 52, c = i - r * 52;
    bufA[r * HSTR + c] = (c < EMB) ? x[(row0 + r) * EMB + c] : 0.0f;
  }
  __syncthreads();

  float* cur = bufA;
  float* nxt = bufB;

  // ---- lin1: [50]->[128], relu ----
  for (int nt = 0; nt < 8; ++nt) {
    const int col = nt * 16 + lm;
    v8f acc = zero8();
#pragma unroll
    for (int s = 0; s < 13; ++s) {
      const int kb = 4 * s + 2 * half;
      v2f a, b;
      a = *(const v2f*)&cur[(m0 + lm) * HSTR + kb];
      if (kb < EMB) { b.x = w1[kb * HID + col]; b.y = (kb + 1 < EMB) ? w1[(kb + 1) * HID + col] : 0.f; }
      else          { b.x = 0.f; b.y = 0.f; }
      acc = wmma_f32(a, b, acc);
    }
    const float bv = b1[col];
#pragma unroll
    for (int r = 0; r < 8; ++r)
      nxt[(m0 + r + 8 * half) * HSTR + col] = fmaxf(acc[r] + bv, 0.f);
  }
  __syncthreads();
  { float* t = cur; cur = nxt; nxt = t; }

  // ---- lin2 x4 (shared weights) + lin3 : [128]->[128] ----
  for (int layer = 0; layer < 5; ++layer) {
    const float* W  = (layer < 4) ? w2 : w3;
    const float* Bs = (layer < 4) ? b2 : b3;
    const bool relu_out = (layer < 4);
    for (int nt = 0; nt < 8; ++nt) {
      const int col = nt * 16 + lm;
      v8f acc = zero8();
#pragma unroll
      for (int s = 0; s < 32; ++s) {
        const int kb = 4 * s + 2 * half;
        v2f a, b;
        a = *(const v2f*)&cur[(m0 + lm) * HSTR + kb];
        b.x = W[kb * HID + col];
        b.y = W[(kb + 1) * HID + col];
        acc = wmma_f32(a, b, acc);
      }
      const float bv = Bs[col];
#pragma unroll
      for (int r = 0; r < 8; ++r) {
        float v = acc[r] + bv;
        nxt[(m0 + r + 8 * half) * HSTR + col] = relu_out ? fmaxf(v, 0.f) : v;
      }
    }
    __syncthreads();
    { float* t = cur; cur = nxt; nxt = t; }
  }

  // ---- 2 residual blocks: h = h + W2(relu(W1(relu(h)))) ----
  for (int i = 0; i < 2; ++i) {
    const float* Rw1 = rw1 + i * HID * RESH;
    const float* Rb1 = rb1 + i * RESH;
    const float* Rw2 = rw2 + i * RESH * HID;
    const float* Rb2 = rb2 + i * HID;

    // u = relu(relu(h) @ Rw1 + Rb1)  : N = 32
    for (int nt = 0; nt < 2; ++nt) {
      const int col = nt * 16 + lm;
      v8f acc = zero8();
#pragma unroll
      for (int s = 0; s < 32; ++s) {
        const int kb = 4 * s + 2 * half;
        v2f a, b;
        a.x = fmaxf(cur[(m0 + lm) * HSTR + kb], 0.f);
        a.y = fmaxf(cur[(m0 + lm) * HSTR + kb + 1], 0.f);
        b.x = Rw1[kb * RESH + col];
        b.y = Rw1[(kb + 1) * RESH + col];
        acc = wmma_f32(a, b, acc);
      }
      const float bv = Rb1[col];
#pragma unroll
      for (int r = 0; r < 8; ++r)
        bufU[(m0 + r + 8 * half) * USTR + col] = fmaxf(acc[r] + bv, 0.f);
    }
    __syncthreads();

    // h_new = h + (u @ Rw2 + Rb2)   : K = 32
    for (int nt = 0; nt < 8; ++nt) {
      const int col = nt * 16 + lm;
      v8f acc = zero8();
#pragma unroll
      for (int s = 0; s < 8; ++s) {
        const int kb = 4 * s + 2 * half;
        v2f a, b;
        a = *(const v2f*)&bufU[(m0 + lm) * USTR + kb];
        b.x = Rw2[kb * HID + col];
        b.y = Rw2[(kb + 1) * HID + col];
        acc = wmma_f32(a, b, acc);
      }
      const float bv = Rb2[col];
#pragma unroll
      for (int r = 0; r < 8; ++r) {
        const int rr = m0 + r + 8 * half;
        nxt[rr * HSTR + col] = cur[rr * HSTR + col] + acc[r] + bv;
      }
    }
    __syncthreads();
    { float* t = cur; cur = nxt; nxt = t; }
  }

  // ---- pre-VQ: z = relu(h) @ pw + pb : N = 250 (16 tiles, last partial) ----
  for (int nt = 0; nt < 16; ++nt) {
    const int col = nt * 16 + lm;
    const bool cok = (col < ZLEN);
    v8f acc = zero8();
#pragma unroll
    for (int s = 0; s < 32; ++s) {
      const int kb = 4 * s + 2 * half;
      v2f a, b;
      a.x = fmaxf(cur[(m0 + lm) * HSTR + kb], 0.f);
      a.y = fmaxf(cur[(m0 + lm) * HSTR + kb + 1], 0.f);
      b.x = cok ? pw[kb * ZLEN + col] : 0.f;
      b.y = cok ? pw[(kb + 1) * ZLEN + col] : 0.f;
      acc = wmma_f32(a, b, acc);
    }
    if (cok) {
      const float bv = pb[col];
#pragma unroll
      for (int r = 0; r < 8; ++r)
        zbuf[(row0 + m0 + r + 8 * half) * ZLEN + col] = acc[r] + bv;
    }
  }
}

// =====================================================================
// codebook row norms: cnorm[n] = sum_k c[n][k]^2
// =====================================================================
__global__ void cnorm_kernel(const float* __restrict__ cb, float* __restrict__ cnorm) {
  const int n = blockIdx.x * 256 + threadIdx.x;
  if (n < KCB) {
    float s = 0.f;
#pragma unroll
    for (int k = 0; k < EMB; ++k) { const float v = cb[n * EMB + k]; s += v * v; }
    cnorm[n] = s;
  }
}

// =====================================================================
// VQ argmin: per wave, 16 tokens vs all 10000 codes via f32 WMMA.
// Codebook tiles (16 codes x 50) are double-buffered in LDS via
// GLOBAL_LOAD_ASYNC_TO_LDS_B64 and shared by all 8 waves of the WG.
// dist_eff = ||c||^2 - 2 * <z, c>  (row-norm dropped: argmin-invariant)
// =====================================================================
__global__ __launch_bounds__(256) void vq_kernel(
    const float* __restrict__ zbuf, const float* __restrict__ cb,
    const float* __restrict__ cnorm, int* __restrict__ idxOut)
{
  __shared__ float cbuf[2][16 * CSTR];   // 2 x 3328 B staged codebook tiles

  const int tid  = threadIdx.x;
  const int lane = tid & 31;
  const int wave = tid >> 5;
  const int half = lane >> 4;
  const int lm   = lane & 15;
  const int t0   = (blockIdx.x * 8 + wave) * 16;   // 160 WGs * 8 waves * 16 = 20480

  // zero pad columns 50/51 of both buffers (async fill never touches them)
  if (tid < 32) {
    const int b = tid >> 4, c = tid & 15;
    cbuf[b][c * CSTR + 50] = 0.f;
    cbuf[b][c * CSTR + 51] = 0.f;
  }

  // A fragments for this wave's 16 tokens, K padded 50 -> 52
  v2f af[13];
  {
    const float* zp = &zbuf[(size_t)(t0 + lm) * EMB];
#pragma unroll
    for (int s = 0; s < 13; ++s) {
      const int kb = 4 * s + 2 * half;
      if (kb < EMB) af[s] = *(const v2f*)(zp + kb);      // kb <= 48, 8B aligned
      else          { af[s].x = 0.f; af[s].y = 0.f; }    // kb == 50 (pad)
    }
  }

  float best[8];
  int   bidx[8];
#pragma unroll
  for (int r = 0; r < 8; ++r) { best[r] = 3.402823e38f; bidx[r] = 0; }

  const unsigned ldsBase0 = (unsigned)(uintptr_t)&cbuf[0][0];
  const unsigned ldsBase1 = (unsigned)(uintptr_t)&cbuf[1][0];
  __syncthreads();                                  // pad zeros visible

  // prologue: fill buffer 0 with tile 0
  stage_tile_async(cb, ldsBase0, 0, tid);

  for (int nt = 0; nt < KCB / 16; ++nt) {           // 625 column tiles
    const int p = nt & 1;
    wait_asynccnt0();                               // my fills of cbuf[p] done
    __syncthreads();                                // everyone's fills done; prev compute done
    if (nt + 1 < KCB / 16)
      stage_tile_async(cb, p ? ldsBase0 : ldsBase1, (nt + 1) * 16, tid);

    const int col = nt * 16 + lm;
    const float* bp = &cbuf[p][lm * CSTR];
    v8f acc = zero8();
#pragma unroll
    for (int s = 0; s < 13; ++s) {
      const int kb = 4 * s + 2 * half;
      const v2f b = *(const v2f*)(bp + kb);         // conflict-free ds_load_b64
      acc = wmma_f32(af[s], b, acc);
    }
    const float cn = cnorm[col];
#pragma unroll
    for (int r = 0; r < 8; ++r) {
      const float d = cn - 2.0f * acc[r];
      if (d < best[r] || (d == best[r] && col < bidx[r])) { best[r] = d; bidx[r] = col; }
    }
  }

  // cross-lane argmin within each 16-lane half (rows 0-7 | 8-15)
#pragma unroll
  for (int r = 0; r < 8; ++r) {
    for (int off = 1; off < 16; off <<= 1) {
      const float ob = __shfl_xor(best[r], off, 16);
      const int   oi = __shfl_xor(bidx[r], off, 16);
      if (ob < best[r] || (ob == best[r] && oi < bidx[r])) { best[r] = ob; bidx[r] = oi; }
    }
  }
  if (lm == 0) {
#pragma unroll
    for (int r = 0; r < 8; ++r) idxOut[t0 + r + 8 * half] = bidx[r];
  }
}

// =====================================================================
// gather quantized (== definition output), squared-error partials
// =====================================================================
__global__ __launch_bounds__(256) void gather_kernel(
    const float* __restrict__ cb, const int* __restrict__ idx,
    const float* __restrict__ zbuf, float* __restrict__ defOut,
    float* __restrict__ partials)
{
  __shared__ float red[256];
  const int e = blockIdx.x * 256 + threadIdx.x;    // 4000 blocks * 256 == TOK*EMB
  const int t = e / EMB, k = e - t * EMB;
  const float q = cb[(size_t)idx[t] * EMB + k];
  const float z = zbuf[e];
  defOut[e] = q;                                   // q_st forward value == quantized
  const float d = q - z;
  red[threadIdx.x] = d * d;
  __syncthreads();
  for (int off = 128; off > 0; off >>= 1) {
    if (threadIdx.x < off) red[threadIdx.x] += red[threadIdx.x + off];
    __syncthreads();
  }
  if (threadIdx.x == 0) partials[blockIdx.x] = red[0];
}

// recon[b][k] = mean_j definition[b][j][k]
__global__ void recon_kernel(const float* __restrict__ defOut, float* __restrict__ reconOut) {
  const int g = blockIdx.x * 256 + threadIdx.x;    // 800 blocks * 256 == BATCH*EMB
  const int b = g / EMB, k = g - b * EMB;
  float s = 0.f;
#pragma unroll
  for (int j = 0; j < DEF; ++j) s += defOut[b * ZLEN + j * EMB + k];
  reconOut[g] = s * (1.0f / DEF);
}

__global__ void zero_counts_kernel(int* __restrict__ counts) {
  const int n = blockIdx.x * 256 + threadIdx.x;
  if (n < KCB) counts[n] = 0;
}

__global__ void hist_kernel(const int* __restrict__ idx, int* __restrict__ counts) {
  const int t = blockIdx.x * 256 + threadIdx.x;
  if (t < TOK) atomicAdd(&counts[idx[t]], 1);
}

// loss + perplexity, fixed-order deterministic reductions
__global__ __launch_bounds__(256) void finalize_kernel(
    const float* __restrict__ partials, const int* __restrict__ counts,
    float* __restrict__ outLoss, float* __restrict__ outPpl)
{
  __shared__ float red[256];
  const int tid = threadIdx.x;

  float s = 0.f;
  for (int i = tid; i < 4000; i += 256) s += partials[i];
  red[tid] = s; __syncthreads();
  for (int off = 128; off > 0; off >>= 1) {
    if (tid < off) red[tid] += red[tid + off];
    __syncthreads();
  }
  if (tid == 0) *outLoss = 0.25f * red[0] / (float)(BATCH * ZLEN);
  __syncthreads();

  float h = 0.f;
  for (int n = tid; n < KCB; n += 256) {
    const float p = (float)counts[n] / (float)TOK;
    h += p * logf(p + 1e-10f);
  }
  red[tid] = h; __syncthreads();
  for (int off = 128; off > 0; off >>= 1) {
    if (tid < off) red[tid] += red[tid + off];
    __syncthreads();
  }
  if (tid == 0) *outPpl = expf(-red[0]);
}

// =====================================================================
extern "C" void kernel_launch(void* const* d_in, const int* in_sizes, int n_in,
                              void* d_out, int out_size, void* d_ws, size_t ws_size,
                              hipStream_t stream) {
  const float* x   = (const float*)d_in[0];
  const float* w1  = (const float*)d_in[1];
  const float* b1  = (const float*)d_in[2];
  const float* w2  = (const float*)d_in[3];
  const float* b2  = (const float*)d_in[4];
  const float* w3  = (const float*)d_in[5];
  const float* b3  = (const float*)d_in[6];
  const float* rw1 = (const float*)d_in[7];
  const float* rb1 = (const float*)d_in[8];
  const float* rw2 = (const float*)d_in[9];
  const float* rb2 = (const float*)d_in[10];
  const float* pw  = (const float*)d_in[11];
  const float* pb  = (const float*)d_in[12];
  const float* cb  = (const float*)d_in[13];

  float* out      = (float*)d_out;
  float* outLoss  = out;                         // [0]
  float* outRecon = out + 1;                     // [1 .. 1+204800)
  float* outDef   = out + 1 + BATCH * EMB;       // [204801 .. +1024000)
  float* outPpl   = out + 1 + BATCH * EMB + TOK * EMB;  // [1228801]

  float* wsf      = (float*)d_ws;
  float* zbuf     = wsf;                         // 1,024,000 f
  float* cnorm    = wsf + TOK * EMB;             //    10,000 f
  float* partials = cnorm + KCB;                 //     4,000 f
  int*   counts   = (int*)(partials + 4000);     //    10,000 i
  int*   idx      = counts + KCB;                //    20,480 i

  const size_t enc_lds = (size_t)(2 * 128 * HSTR + 128 * USTR) * sizeof(float);

  enc_kernel<<<BATCH / 128, 256, enc_lds, stream>>>(x, w1, b1, w2, b2, w3, b3,
                                                    rw1, rb1, rw2, rb2, pw, pb, zbuf);
  cnorm_kernel<<<(KCB + 255) / 256, 256, 0, stream>>>(cb, cnorm);
  vq_kernel<<<TOK / 128, 256, 0, stream>>>(zbuf, cb, cnorm, idx);
  gather_kernel<<<(TOK * EMB) / 256, 256, 0, stream>>>(cb, idx, zbuf, outDef, partials);
  recon_kernel<<<(BATCH * EMB) / 256, 256, 0, stream>>>(outDef, outRecon);
  zero_counts_kernel<<<(KCB + 255) / 256, 256, 0, stream>>>(counts);
  hist_kernel<<<(TOK + 255) / 256, 256, 0, stream>>>(idx, counts);
  finalize_kernel<<<1, 256, 0, stream>>>(partials, counts, outLoss, outPpl);
}